// GraphAttentionLayer_85272280695285
// MI455X (gfx1250) — compile-verified
//
#include <hip/hip_runtime.h>
#include <cmath>

typedef __attribute__((ext_vector_type(2))) float v2f;
typedef __attribute__((ext_vector_type(8))) float v8f;

#define BN   2048   // nodes
#define INF_ 512    // in features
#define OUTF 512    // out features
#define NH   8      // heads
#define HD   64     // dim per head
#define TI   16     // i-tile rows per workgroup (kernel C)
#define CJ   512    // j chunk (kernel C)

// -------------------------------------------------------------------------
// Kernel A: h = x @ W^T + b  (fp32 WMMA 16x16x4)
// 1024 waves: each computes rows m0..m0+15 x cols n0..n0+63 (4 N-tiles),
// reusing one A fragment per K-step across 4 WMMAs.
// A-frag layout: lane = mi + 16*kk holds A[mi][2kk], A[mi][2kk+1]
// B-frag layout: lane = ni + 16*kk holds B[2kk][ni], B[2kk+1][ni]  (B = W^T)
// -------------------------------------------------------------------------
__global__ __launch_bounds__(128)
void gat_linear_wmma(const float* __restrict__ x, const float* __restrict__ W,
                     const float* __restrict__ bias, float* __restrict__ h)
{
    const int lane = threadIdx.x & 31;
    const int wave = threadIdx.x >> 5;
    const int gw   = blockIdx.x * 4 + wave;        // 0..1023
    const int m0   = (gw & 127) << 4;              // 128 m-tiles of 16
    const int n0   = (gw >> 7) << 6;               // 8 n-groups of 64
    const int mi   = lane & 15;
    const int kk   = lane >> 4;                    // selects K pair {2kk,2kk+1}

    v8f acc0 = {}, acc1 = {}, acc2 = {}, acc3 = {};

    const float* xr  = x + (size_t)(m0 + mi) * INF_ + 2 * kk;
    const float* wr0 = W + (size_t)(n0 +  0 + mi) * INF_ + 2 * kk;
    const float* wr1 = W + (size_t)(n0 + 16 + mi) * INF_ + 2 * kk;
    const float* wr2 = W + (size_t)(n0 + 32 + mi) * INF_ + 2 * kk;
    const float* wr3 = W + (size_t)(n0 + 48 + mi) * INF_ + 2 * kk;

    for (int k0 = 0; k0 < INF_; k0 += 4) {
        v2f a  = *(const v2f*)(xr  + k0);
        v2f b0 = *(const v2f*)(wr0 + k0);
        v2f b1 = *(const v2f*)(wr1 + k0);
        v2f b2 = *(const v2f*)(wr2 + k0);
        v2f b3 = *(const v2f*)(wr3 + k0);
        acc0 = __builtin_amdgcn_wmma_f32_16x16x4_f32(false, a, false, b0, (short)0, acc0, false, false);
        acc1 = __builtin_amdgcn_wmma_f32_16x16x4_f32(false, a, false, b1, (short)0, acc1, false, false);
        acc2 = __builtin_amdgcn_wmma_f32_16x16x4_f32(false, a, false, b2, (short)0, acc2, false, false);
        acc3 = __builtin_amdgcn_wmma_f32_16x16x4_f32(false, a, false, b3, (short)0, acc3, false, false);
    }

    // bias + store: acc vgpr r <-> row m0 + r + 8*kk, col (n-tile base) + mi
    v8f accs[4] = {acc0, acc1, acc2, acc3};
    #pragma unroll
    for (int t = 0; t < 4; ++t) {
        const int col = n0 + t * 16 + mi;
        const float bv = bias[col];
        #pragma unroll
        for (int r = 0; r < 8; ++r) {
            const int row = m0 + r + 8 * kk;
            h[(size_t)row * OUTF + col] = accs[t][r] + bv;
        }
    }
}

// -------------------------------------------------------------------------
// Kernel B: s_src[b,hd] = sum_d h[b,hd*64+d]*a[d] ; s_dst uses a[64+d]
// -------------------------------------------------------------------------
__global__ __launch_bounds__(256)
void gat_scores(const float* __restrict__ h, const float* __restrict__ a,
                float* __restrict__ s_src, float* __restrict__ s_dst)
{
    const int t = blockIdx.x * 256 + threadIdx.x;   // 0..16383
    if (t >= BN * NH) return;
    const int b  = t >> 3;
    const int hd = t & 7;
    const float* hp = h + (size_t)b * OUTF + hd * HD;
    float s1 = 0.f, s2 = 0.f;
    #pragma unroll 4
    for (int d = 0; d < HD; ++d) {
        const float v = hp[d];
        s1 += v * a[d];
        s2 += v * a[HD + d];
    }
    s_src[t] = s1;
    s_dst[t] = s2;
}

// -------------------------------------------------------------------------
// Kernel C: fused masked-softmax attention + aggregation + ELU.
// grid (128, 8): blockIdx.x = 16-row i-tile, blockIdx.y = head. 128 threads.
// Chunks j by 512: phase1 builds unnormalized exp scores in LDS and the
// softmax denominator; phase2 does h' tile += p~ @ h via fp32 WMMA, one wave
// per 16-col d-tile. 1/denom folded into the final accumulator, then ELU.
// -------------------------------------------------------------------------
__global__ __launch_bounds__(128)
void gat_attn_aggregate(const float* __restrict__ h, const int* __restrict__ adj,
                        const float* __restrict__ s_src, const float* __restrict__ s_dst,
                        float* __restrict__ out)
{
    __shared__ float pbuf[CJ][TI];   // 32 KB, j-major -> conflict-free WMMA A reads
    __shared__ float sdst[CJ];       // 2 KB
    __shared__ float ssrc[TI];
    __shared__ float red[128];
    __shared__ float invd[TI];

    const int i0   = blockIdx.x * TI;
    const int hd   = blockIdx.y;
    const int t    = threadIdx.x;
    const int lane = t & 31;
    const int wave = t >> 5;
    const int mi   = lane & 15;
    const int kk   = lane >> 4;
    const int d0   = wave * 16;      // this wave's 16 output columns

    if (t < TI) ssrc[t] = s_src[(size_t)(i0 + t) * NH + hd];

    const int r = t >> 3;            // row 0..15 (phase 1)
    const int s = t & 7;             // j-slice 0..7 (phase 1)
    float part = 0.f;                // partial softmax denominator (row r)
    v8f acc = {};

    for (int jb = 0; jb < BN; jb += CJ) {
        __syncthreads();             // previous chunk's phase2 done with LDS
        for (int q = t; q < CJ; q += 128)
            sdst[q] = s_dst[(size_t)(jb + q) * NH + hd];
        __syncthreads();

        // ---- phase 1: p~ = adj ? exp(leaky(s_src+s_dst)) : 0 ----
        const int*  adjrow = adj + (size_t)(i0 + r) * BN + jb;
        const float sr = ssrc[r];
        for (int jj = s; jj < CJ; jj += 8) {
            float e = sr + sdst[jj];
            e = (e >= 0.f) ? e : 0.2f * e;
            const float p = adjrow[jj] ? __expf(e) : 0.f;
            pbuf[jj][r] = p;
            part += p;
        }
        __syncthreads();

        // ---- phase 2: acc += p~ @ h  (WMMA f32 16x16x4, K over j) ----
        const float* hb = h + (size_t)(jb + 2 * kk) * OUTF + hd * HD + d0 + mi;
        for (int j0 = 0; j0 < CJ; j0 += 4) {
            v2f av;
            av.x = pbuf[j0 + 2 * kk][mi];
            av.y = pbuf[j0 + 2 * kk + 1][mi];
            v2f bv;
            bv.x = hb[(size_t)(j0)     * OUTF];
            bv.y = hb[(size_t)(j0 + 1) * OUTF];
            acc = __builtin_amdgcn_wmma_f32_16x16x4_f32(false, av, false, bv, (short)0, acc, false, false);
        }
    }

    // ---- softmax denominator reduction (8 partials per row) ----
    red[t] = part;
    __syncthreads();
    if (t < TI) {
        float ssum = 0.f;
        #pragma unroll
        for (int q = 0; q < 8; ++q) ssum += red[t * 8 + q];
        invd[t] = (ssum > 0.f) ? (1.f / ssum) : 0.f;
    }
    __syncthreads();

    // ---- scale by 1/denom, ELU, store ----
    #pragma unroll
    for (int rr = 0; rr < 8; ++rr) {
        const int row = rr + 8 * kk;
        float v = acc[rr] * invd[row];
        v = (v > 0.f) ? v : expm1f(v);
        out[(size_t)(i0 + row) * OUTF + hd * HD + d0 + mi] = v;
    }
}

// -------------------------------------------------------------------------
extern "C" void kernel_launch(void* const* d_in, const int* in_sizes, int n_in,
                              void* d_out, int out_size, void* d_ws, size_t ws_size,
                              hipStream_t stream)
{
    const float* x   = (const float*)d_in[0];
    const int*   adj = (const int*)  d_in[1];
    const float* Ww  = (const float*)d_in[2];
    const float* Wb  = (const float*)d_in[3];
    const float* a   = (const float*)d_in[4];
    float* out = (float*)d_out;

    float* h     = (float*)d_ws;                 // 2048*512 f32 = 4 MB
    float* s_src = h + (size_t)BN * OUTF;        // 2048*8
    float* s_dst = s_src + (size_t)BN * NH;      // 2048*8

    gat_linear_wmma<<<256, 128, 0, stream>>>(x, Ww, Wb, h);
    gat_scores<<<(BN * NH) / 256, 256, 0, stream>>>(h, a, s_src, s_dst);
    dim3 grid(BN / TI, NH);
    gat_attn_aggregate<<<grid, 128, 0, stream>>>(h, adj, s_src, s_dst, out);
}